// GraphConvDown_22488448761962
// MI455X (gfx1250) — compile-verified
//
#include <hip/hip_runtime.h>
#include <hip/hip_bf16.h>

typedef __attribute__((ext_vector_type(2))) float v2f;
typedef __attribute__((ext_vector_type(8))) float v8f;

#define C_IN    32
#define C_OUT   64
#define K1      35   // C_IN + 3 (rel xyz)
#define K1PAD   36
#define W1PITCH 38   // even (8B-aligned pairs), 38*lane % 64 distinct over 16 lanes
#define W2PITCH 66   // even, 66*lane % 64 = 2*lane -> distinct bank pairs
#define HPITCH  66   // even, conflict-free for row-indexed loads

// ---------------------------------------------------------------------------
// Kernel 1: new_bxyz = point_bxyz[sample_idx]   ([M,4] float4 gather-copy)
// ---------------------------------------------------------------------------
__global__ void sample_gather(const float* __restrict__ bxyz,
                              const int*   __restrict__ sidx,
                              float*       __restrict__ new_bxyz,
                              int m)
{
    int i = blockIdx.x * blockDim.x + threadIdx.x;
    if (i < m) {
        int p = sidx[i];
        const float4* src = (const float4*)bxyz;
        float4*       dst = (float4*)new_bxyz;
        dst[i] = src[p];
    }
}

// ---------------------------------------------------------------------------
// Kernel 2: per-edge 2-layer MLP via fp32 WMMA + scatter-max (int atomicMax
// on non-negative float bit patterns). One wave32 per 16-edge tile.
// Weights stored TRANSPOSED in LDS so every WMMA B-pair is one ds_load_b64
// into an aligned VGPR pair (kills the v_mov packing traffic).
// ---------------------------------------------------------------------------
__launch_bounds__(256)
__global__ void edge_mlp_wmma(const float* __restrict__ bxyz,
                              const float* __restrict__ feat,
                              const int*   __restrict__ e_point,
                              const int*   __restrict__ e_new,
                              const float* __restrict__ W1,
                              const float* __restrict__ b1,
                              const float* __restrict__ W2,
                              const float* __restrict__ b2,
                              const float* __restrict__ new_bxyz,
                              float*       __restrict__ agg,
                              int n_edges, int n_tiles)
{
    __shared__ float sW1t[C_OUT * W1PITCH];   // [col][k], k=35 zero-padded:  9728 B
    __shared__ float sW2t[C_OUT * W2PITCH];   // [col][k]:                   16896 B
    __shared__ float sH[8][16 * HPITCH];      // per-wave h staging:         33792 B

    const int tid = threadIdx.x;
    // transpose weights into LDS (cooperative, 256 threads)
    for (int i = tid; i < C_OUT * K1PAD; i += 256) {
        const int c = i / K1PAD, k = i - c * K1PAD;
        sW1t[c * W1PITCH + k] = (k < K1) ? W1[k * C_OUT + c] : 0.0f;
    }
    for (int i = tid; i < C_OUT * C_OUT; i += 256) {
        const int c = i >> 6, k = i & 63;
        sW2t[c * W2PITCH + k] = W2[k * C_OUT + c];
    }
    __syncthreads();

    const int wave = tid >> 5;
    const int lane = tid & 31;
    const int tile = blockIdx.x * 8 + wave;
    if (tile >= n_tiles) return;

    const int m    = lane & 15;   // A-layout row index == C/D-layout column index
    const int half = lane >> 4;
    int e = tile * 16 + m;
    if (e >= n_edges) e = n_edges - 1;   // duplicate edge: idempotent under max
    const int pe = e_point[e];
    const int ne = e_new[e];

    // relative position (xyz, cols 1..3)
    float rel[3];
#pragma unroll
    for (int j = 0; j < 3; ++j)
        rel[j] = bxyz[pe * 4 + 1 + j] - new_bxyz[ne * 4 + 1 + j];

    const float* frow = feat + (long)pe * C_IN;
    const int col = m;   // column within each 16-wide n-block

    // ---------------- layer 1: D = A(16x35) @ W1(35x64) + b1 ----------------
    v8f c1[4];
#pragma unroll
    for (int nb = 0; nb < 4; ++nb) {
        const float bv = b1[nb * 16 + col];
#pragma unroll
        for (int r = 0; r < 8; ++r) c1[nb][r] = bv;
    }

#pragma unroll
    for (int s = 0; s < 9; ++s) {
        const int ka = 4 * s + 2 * half;     // even -> 8B-aligned LDS pairs
        const int kb = ka + 1;
        const float a0 = (ka < C_IN) ? frow[ka] : ((ka < K1) ? rel[ka - C_IN] : 0.0f);
        const float a1 = (kb < C_IN) ? frow[kb] : ((kb < K1) ? rel[kb - C_IN] : 0.0f);
        const v2f a = {a0, a1};
#pragma unroll
        for (int nb = 0; nb < 4; ++nb) {
            const v2f b = *(const v2f*)&sW1t[(nb * 16 + col) * W1PITCH + ka];
            c1[nb] = __builtin_amdgcn_wmma_f32_16x16x4_f32(
                         false, a, false, b, (short)0, c1[nb], false, false);
        }
    }

    // ReLU + transpose-stage h tile through LDS (C/D layout -> A layout)
    float* hrow = sH[wave];
#pragma unroll
    for (int nb = 0; nb < 4; ++nb) {
#pragma unroll
        for (int r = 0; r < 8; ++r) {
            float v = c1[nb][r];
            v = v > 0.0f ? v : 0.0f;
            const int mm = r + 8 * half;
            hrow[mm * HPITCH + nb * 16 + col] = v;
        }
    }

    // ---------------- layer 2: D = h(16x64) @ W2(64x64) + b2 ----------------
    v8f c2[4];
#pragma unroll
    for (int nb = 0; nb < 4; ++nb) {
        const float bv = b2[nb * 16 + col];
#pragma unroll
        for (int r = 0; r < 8; ++r) c2[nb][r] = bv;
    }

#pragma unroll
    for (int s = 0; s < 16; ++s) {
        const int ka = 4 * s + 2 * half;     // even -> aligned v2f load
        const v2f a = *(const v2f*)&hrow[m * HPITCH + ka];
#pragma unroll
        for (int nb = 0; nb < 4; ++nb) {
            const v2f b = *(const v2f*)&sW2t[(nb * 16 + col) * W2PITCH + ka];
            c2[nb] = __builtin_amdgcn_wmma_f32_16x16x4_f32(
                         false, a, false, b, (short)0, c2[nb], false, false);
        }
    }

    // -------- ReLU + scatter-max (non-negative floats: int atomicMax) -------
#pragma unroll
    for (int r = 0; r < 8; ++r) {
        const int mm  = r + 8 * half;
        const int row = __shfl(ne, mm, 32);   // e_new of edge row mm
#pragma unroll
        for (int nb = 0; nb < 4; ++nb) {
            float v = c2[nb][r];
            v = v > 0.0f ? v : 0.0f;
            atomicMax((int*)&agg[(long)row * C_OUT + nb * 16 + col],
                      __float_as_int(v));
        }
    }
}

// ---------------------------------------------------------------------------
extern "C" void kernel_launch(void* const* d_in, const int* in_sizes, int n_in,
                              void* d_out, int out_size, void* d_ws, size_t ws_size,
                              hipStream_t stream)
{
    const float* bxyz    = (const float*)d_in[0];
    const float* feat    = (const float*)d_in[1];
    const int*   sidx    = (const int*)  d_in[2];
    const int*   e_point = (const int*)  d_in[3];
    const int*   e_new   = (const int*)  d_in[4];
    const float* W1      = (const float*)d_in[5];
    const float* b1      = (const float*)d_in[6];
    const float* W2      = (const float*)d_in[7];
    const float* b2      = (const float*)d_in[8];

    const int M = in_sizes[2];        // sample_idx count
    const int E = in_sizes[3];        // edge count

    float* out      = (float*)d_out;
    float* new_bxyz = out;                       // [M,4]
    float* agg      = out + (size_t)M * 4;       // [M,64]

    // agg starts at 0: matches segment_max over ReLU outputs (>=0) and the
    // reference's isfinite()->0 handling of empty segments.
    hipMemsetAsync(agg, 0, (size_t)M * C_OUT * sizeof(float), stream);

    sample_gather<<<(M + 255) / 256, 256, 0, stream>>>(bxyz, sidx, new_bxyz, M);

    const int n_tiles = (E + 15) / 16;
    edge_mlp_wmma<<<(n_tiles + 7) / 8, 256, 0, stream>>>(
        bxyz, feat, e_point, e_new, W1, b1, W2, b2, new_bxyz, agg, E, n_tiles);
}